// MaskRCNN_48120813584721
// MI455X (gfx1250) — compile-verified
//
#include <hip/hip_runtime.h>
#include <stdint.h>

#define POOL 7
#define NCH  256
#define PDW  48   // param dwords per ROI (44 used + pad to 16B multiple)

// ---------------------------------------------------------------------------
// Kernel 1: per-ROI parameter setup (level select + bilinear weights/indices)
// ---------------------------------------------------------------------------
__global__ void roi_setup_kernel(const float* __restrict__ rois,
                                 int* __restrict__ wsp, int nroi)
{
    int r = blockIdx.x * blockDim.x + threadIdx.x;
    if (r >= nroi) return;

    float x1 = rois[4*r+0], y1 = rois[4*r+1];
    float x2 = rois[4*r+2], y2 = rois[4*r+3];

    // level = clip(round(log2(sqrt(area)/224)) + 4, 2, 5)   (round = half-even)
    float area = (y2 - y1) * (x2 - x1);
    float rl   = log2f(sqrtf(fmaxf(area, 0.0f)) / 224.0f);
    float lv   = rintf(rl) + 4.0f;
    lv = fminf(fmaxf(lv, 2.0f), 5.0f);
    int lvl = (int)lv;
    int H   = 1024 >> lvl;              // W == H for every level
    float Hm1 = (float)(H - 1);

    const float inv = 1.0f / 1024.0f;   // exact power of two
    float by1 = y1*inv, by2 = y2*inv, bx1 = x1*inv, bx2 = x2*inv;

    int*   pi = wsp + (size_t)r * PDW;
    float* pf = (float*)pi;
    pi[0] = lvl - 2;                    // 0..3 -> p2..p5
    pi[1] = H;

    #pragma unroll
    for (int j = 0; j < POOL; ++j) {
        float t = (float)j / 6.0f;

        // --- y axis ---
        float yv = (by1 + t * (by2 - by1)) * Hm1;
        float vy = (yv >= 0.0f && yv <= Hm1) ? 1.0f : 0.0f;
        float yc = fminf(fmaxf(yv, 0.0f), Hm1);
        int   y0 = (int)floorf(yc);
        if (y0 > H - 2) y0 = H - 2;     // yc==H-1 -> (y0=H-2, ly=1), identical value
        float ly = yc - (float)y0;
        pi[2  + j] = y0 * H;            // row offset in elements
        pf[9  + j] = (1.0f - ly) * vy;  // mask folded into weights
        pf[16 + j] = ly * vy;

        // --- x axis ---
        float xv = (bx1 + t * (bx2 - bx1)) * Hm1;
        float vx = (xv >= 0.0f && xv <= Hm1) ? 1.0f : 0.0f;
        float xc = fminf(fmaxf(xv, 0.0f), Hm1);
        int   x0 = (int)floorf(xc);
        if (x0 > H - 2) x0 = H - 2;
        float lx = xc - (float)x0;
        pi[23 + j] = x0;
        pf[30 + j] = (1.0f - lx) * vx;
        pf[37 + j] = lx * vx;
    }
    pi[44] = pi[45] = pi[46] = pi[47] = 0;
}

// ---------------------------------------------------------------------------
// Kernel 2: one block per ROI, one lane per channel (8 wave32s / block).
// Params broadcast to the whole block through LDS with an async DMA load.
// ---------------------------------------------------------------------------
__global__ __launch_bounds__(NCH)
void roi_align_kernel(const float* __restrict__ p2, const float* __restrict__ p3,
                      const float* __restrict__ p4, const float* __restrict__ p5,
                      const int* __restrict__ wsp, float* __restrict__ out)
{
    __shared__ __align__(16) int spi[PDW];

    const int roi = blockIdx.x;
    const int c   = threadIdx.x;
    const int* gp = wsp + (size_t)roi * PDW;

    // Stage 192B of ROI params into LDS: 12 lanes x b128 async DMA.
    if (threadIdx.x < PDW/4) {
        unsigned lds_addr = (unsigned)(uintptr_t)&spi[threadIdx.x * 4]; // low 32b of flat = LDS offset
        const int* ga = gp + threadIdx.x * 4;
        asm volatile("global_load_async_to_lds_b128 %0, %1, off"
                     :: "v"(lds_addr), "v"(ga) : "memory");
    }
    asm volatile("s_wait_asynccnt 0" ::: "memory");
    __syncthreads();

    const float* spf = (const float*)spi;
    const int lvl = spi[0];
    const int W   = spi[1];
    const float* fm = (lvl == 0) ? p2 : (lvl == 1) ? p3 : (lvl == 2) ? p4 : p5;

    const float* fmc  = fm + (size_t)c * W * W;                     // image 0, channel c
    float*       outr = out + ((size_t)roi * NCH + c) * (POOL*POOL);

    #pragma unroll
    for (int py = 0; py < POOL; ++py) {
        const int   yo  = spi[2  + py];
        const float wy0 = spf[9  + py];
        const float wy1 = spf[16 + py];
        const float* r0 = fmc + yo;
        const float* r1 = r0 + W;

        // overlap gather latency: prefetch next grid row (global_prefetch_b8)
        if (py < POOL - 1) {
            int yn = spi[3 + py];
            __builtin_prefetch(fmc + yn +     spi[23], 0, 0);
            __builtin_prefetch(fmc + yn + W + spi[23], 0, 0);
        }

        #pragma unroll
        for (int px = 0; px < POOL; ++px) {
            const int   x0  = spi[23 + px];
            const float wx0 = spf[30 + px];
            const float wx1 = spf[37 + px];
            float tl = r0[x0],     tr = r0[x0 + 1];
            float bl = r1[x0],     br = r1[x0 + 1];
            float top = tl * wx0 + tr * wx1;
            float bot = bl * wx0 + br * wx1;
            outr[py*POOL + px] = top * wy0 + bot * wy1;
        }
    }
}

// ---------------------------------------------------------------------------
extern "C" void kernel_launch(void* const* d_in, const int* in_sizes, int n_in,
                              void* d_out, int out_size, void* d_ws, size_t ws_size,
                              hipStream_t stream)
{
    const float* p2   = (const float*)d_in[0];
    const float* p3   = (const float*)d_in[1];
    const float* p4   = (const float*)d_in[2];
    const float* p5   = (const float*)d_in[3];
    const float* rois = (const float*)d_in[4];

    const int nroi = in_sizes[4] / 4;           // B*R = 2000
    int* wsp = (int*)d_ws;                       // needs nroi*PDW*4 = 384 KB

    roi_setup_kernel<<<(nroi + 255) / 256, 256, 0, stream>>>(rois, wsp, nroi);
    roi_align_kernel<<<nroi, NCH, 0, stream>>>(p2, p3, p4, p5, wsp, (float*)d_out);
}